// RNNTorch_42812234006864
// MI455X (gfx1250) — compile-verified
//
#include <hip/hip_runtime.h>
#include <hip/hip_bf16.h>
#include <math.h>

// Problem constants (match reference)
#define BQ   4
#define SQ   4096
#define DQ   2048          // D_IN == D_ST == D_OUT
#define NH   16            // heads
#define HH   128           // head dim
#define MROWS (BQ * SQ)    // 16384

typedef __attribute__((ext_vector_type(2))) float v2f;
typedef __attribute__((ext_vector_type(4))) float v4f;
typedef __attribute__((ext_vector_type(8))) float v8f;

#define WMMA_F32(a, b, c) \
    __builtin_amdgcn_wmma_f32_16x16x4_f32(false, (a), false, (b), (short)0, (c), false, false)

// ---- CDNA5 async global->LDS copy (ASYNCcnt-tracked), with sync fallback ----
#if defined(__has_builtin)
#  if __has_builtin(__builtin_amdgcn_global_load_async_to_lds_b128) && \
      __has_builtin(__builtin_amdgcn_s_wait_asynccnt)
#    define HAVE_ASYNC_LDS 1
#  endif
#endif

#ifdef HAVE_ASYNC_LDS
// builtin expects: (vector_size(16) int __attribute__((address_space(1)))* src,
//                   vector_size(16) int __attribute__((address_space(3)))* dst,
//                   imm offset, imm cpol)
typedef int vsi4 __attribute__((vector_size(16)));
typedef __attribute__((address_space(1))) vsi4 as1_vsi4;
typedef __attribute__((address_space(3))) vsi4 as3_vsi4;

__device__ __forceinline__ void async_cp16(const float* g, float* l) {
    // per-lane 16B copy from the global address g to the LDS address l, tracked by ASYNCcnt
    __builtin_amdgcn_global_load_async_to_lds_b128(
        (as1_vsi4*)(uintptr_t)g,
        (as3_vsi4*)(uint32_t)(uintptr_t)l,
        0, 0);
}
#endif

// ---------------------------------------------------------------------------
// GEMM: C[M,N] = A[M,K] @ W[N,K]^T (+ bias[N]), fp32, M=16384, N=K=2048.
// 128x128 tile per 256-thread workgroup (8 waves). Wave w computes rows
// 16w..16w+15 x all 128 cols (8 WMMA f32 16x16x4 accumulators).
// Double-buffered LDS staging via async-to-LDS (BK=32 per stage).
// ---------------------------------------------------------------------------
#define BM 128
#define BN 128
#define BK 32
#define LDP 36   // padded LDS row stride (floats): conflict-free, b64/b128 aligned
#define NCHUNK (DQ / BK)   // 64

__global__ __launch_bounds__(256)
void gemm_f32_wmma(const float* __restrict__ A,
                   const float* __restrict__ W,     // [N,K] row-major
                   const float* __restrict__ bias,  // may be null
                   float* __restrict__ C) {
    __shared__ float As[2][BM * LDP];
    __shared__ float Bs[2][BN * LDP];

    const int tid  = threadIdx.x;
    const int w    = tid >> 5;        // wave 0..7
    const int lane = tid & 31;
    const int lm   = lane & 15;       // lane % 16
    const int lh   = lane >> 4;       // lane / 16
    const int mBase = blockIdx.y * BM;
    const int nBase = blockIdx.x * BN;

    v8f acc[8] = {};

    // per-thread staging assignment: row = tid/2, 16-col half = (tid&1)*16
    const int ldRow = tid >> 1;
    const int ldCol = (tid & 1) * 16;
    const float* gA = A + (size_t)(mBase + ldRow) * DQ + ldCol;
    const float* gB = W + (size_t)(nBase + ldRow) * DQ + ldCol;
    const int ldsOff = ldRow * LDP + ldCol;

#ifdef HAVE_ASYNC_LDS
    // prologue: stage chunk 0 into buffer 0
#pragma unroll
    for (int i = 0; i < 4; ++i) {
        async_cp16(gA + 4 * i, &As[0][ldsOff + 4 * i]);
        async_cp16(gB + 4 * i, &Bs[0][ldsOff + 4 * i]);
    }
    for (int c = 0; c < NCHUNK; ++c) {
        const int cur = c & 1;
        if (c + 1 < NCHUNK) {
            const int k1 = (c + 1) * BK;
#pragma unroll
            for (int i = 0; i < 4; ++i) {
                async_cp16(gA + k1 + 4 * i, &As[cur ^ 1][ldsOff + 4 * i]);
                async_cp16(gB + k1 + 4 * i, &Bs[cur ^ 1][ldsOff + 4 * i]);
            }
            __builtin_amdgcn_s_wait_asynccnt(8);  // chunk c done; c+1 in flight
        } else {
            __builtin_amdgcn_s_wait_asynccnt(0);
        }
        __syncthreads();
#pragma unroll
        for (int kk = 0; kk < BK; kk += 4) {
            v2f a = *(const v2f*)(&As[cur][(16 * w + lm) * LDP + kk + 2 * lh]);
#pragma unroll
            for (int j = 0; j < 8; ++j) {
                v2f b = *(const v2f*)(&Bs[cur][(16 * j + lm) * LDP + kk + 2 * lh]);
                acc[j] = WMMA_F32(a, b, acc[j]);
            }
        }
        __syncthreads();  // all waves done reading buf[cur] -> reusable
    }
#else
    // fallback: synchronous VGPR staging, single-buffer semantics
    for (int c = 0; c < NCHUNK; ++c) {
        const int k0 = c * BK;
        __syncthreads();
#pragma unroll
        for (int i = 0; i < 4; ++i) {
            *(v4f*)(&As[0][ldsOff + 4 * i]) = *(const v4f*)(gA + k0 + 4 * i);
            *(v4f*)(&Bs[0][ldsOff + 4 * i]) = *(const v4f*)(gB + k0 + 4 * i);
        }
        __syncthreads();
#pragma unroll
        for (int kk = 0; kk < BK; kk += 4) {
            v2f a = *(const v2f*)(&As[0][(16 * w + lm) * LDP + kk + 2 * lh]);
#pragma unroll
            for (int j = 0; j < 8; ++j) {
                v2f b = *(const v2f*)(&Bs[0][(16 * j + lm) * LDP + kk + 2 * lh]);
                acc[j] = WMMA_F32(a, b, acc[j]);
            }
        }
    }
#endif

    // Epilogue: C layout vgpr v -> row v + 8*lh, col lm
#pragma unroll
    for (int j = 0; j < 8; ++j) {
        const int col = nBase + 16 * j + lm;
        const float bv = bias ? bias[col] : 0.0f;
#pragma unroll
        for (int v = 0; v < 8; ++v) {
            const int row = mBase + 16 * w + v + 8 * lh;
            C[(size_t)row * DQ + col] = acc[j][v] + bv;
        }
    }
}

// ---------------------------------------------------------------------------
// Recurrent scan: one workgroup per head n. W[n] (128x128) transposed into
// LDS (Ws[c][k], padded) so B fragments are single b64 loads. Padded 16x128
// state tile in LDS (rows 4..15 stay zero since B=4). Per step: 32 chained
// f32 WMMAs per wave; lanes 0..15 apply tanh(acc + x_t) to rows 0..3,
// update state, write ys in place over h.
// ---------------------------------------------------------------------------
#define STP 132  // padded state row stride (floats)
#define WSP 132  // padded Ws row stride (floats)

__global__ __launch_bounds__(256)
void rnn_scan_wmma(float* __restrict__ h,          // [B*S, 2048], in/out (ys)
                   const float* __restrict__ sw) { // [N, H, H]
    __shared__ float Ws[HH * WSP];  // Ws[c*WSP + k] = W[k][c]  (transposed)
    __shared__ float St[16 * STP];  // padded state, rows 4..15 stay zero

    const int n    = blockIdx.x;
    const int tid  = threadIdx.x;
    const int w    = tid >> 5;
    const int lane = tid & 31;
    const int lm   = lane & 15;
    const int lh   = lane >> 4;

    // stage W[n] transposed: thread covers row k = tid/2, 64-col half
    {
        const float* Wn = sw + (size_t)n * HH * HH;
        const int k     = tid >> 1;
        const int cbase = (tid & 1) * 64;
#pragma unroll
        for (int i = 0; i < 64; i += 4) {
            v4f v = *(const v4f*)(Wn + k * HH + cbase + i);
            Ws[(cbase + i + 0) * WSP + k] = v.x;
            Ws[(cbase + i + 1) * WSP + k] = v.y;
            Ws[(cbase + i + 2) * WSP + k] = v.z;
            Ws[(cbase + i + 3) * WSP + k] = v.w;
        }
    }
    // zero state
    for (int idx = tid; idx < 16 * STP; idx += 256) St[idx] = 0.0f;
    __syncthreads();

    const int col = n * HH + 16 * w + lm;  // this lane's output column (lanes 0..15)

    for (int t = 0; t < SQ; ++t) {
        // prefetch x_t rows 0..3 (only meaningful for lanes 0..15)
        float xin[4];
        if (lh == 0) {
#pragma unroll
            for (int v = 0; v < 4; ++v)
                xin[v] = h[((size_t)v * SQ + t) * DQ + col];
        }

        v8f acc = {};
#pragma unroll
        for (int kk = 0; kk < HH; kk += 4) {
            v2f a = *(const v2f*)(&St[lm * STP + kk + 2 * lh]);
            v2f b = *(const v2f*)(&Ws[(16 * w + lm) * WSP + kk + 2 * lh]);
            acc = WMMA_F32(a, b, acc);
        }

        __syncthreads();   // all waves done reading previous state
        if (lh == 0) {
#pragma unroll
            for (int v = 0; v < 4; ++v) {           // rows 0..3 == batch index
                const float val = tanhf(acc[v] + xin[v]);
                h[((size_t)v * SQ + t) * DQ + col] = val;   // ys in place
                St[v * STP + 16 * w + lm] = val;
            }
        }
        __syncthreads();   // state update visible before next step's reads
    }
}

// ---------------------------------------------------------------------------
extern "C" void kernel_launch(void* const* d_in, const int* in_sizes, int n_in,
                              void* d_out, int out_size, void* d_ws, size_t ws_size,
                              hipStream_t stream) {
    const float* x     = (const float*)d_in[0];  // [B,S,D_IN]
    const float* w_in  = (const float*)d_in[1];  // [D_ST, D_IN]
    const float* b_in  = (const float*)d_in[2];  // [D_ST]
    const float* sw    = (const float*)d_in[3];  // [N,H,H]
    const float* w_out = (const float*)d_in[4];  // [D_OUT, D_ST]
    float* out = (float*)d_out;
    float* h   = (float*)d_ws;                   // [B*S, D_ST] scratch (134 MB)

    dim3 blk(256);
    dim3 gGemm(DQ / BN, MROWS / BM);             // (16, 128)

    // 1) h = x @ w_in^T + b_in
    gemm_f32_wmma<<<gGemm, blk, 0, stream>>>(x, w_in, b_in, h);
    // 2) sequential tanh recurrence per (b, head); ys overwrites h in place
    rnn_scan_wmma<<<dim3(NH), blk, 0, stream>>>(h, sw);
    // 3) out = ys @ w_out^T
    gemm_f32_wmma<<<gGemm, blk, 0, stream>>>(h, w_out, nullptr, out);
}